// LinAminoToAtomSmall_51719996178670
// MI455X (gfx1250) — compile-verified
//
#include <hip/hip_runtime.h>
#include <cstdint>

// Problem constants (from reference)
#define BATCH   16
#define AMINOS  1024
#define ATOMS   8192
#define DIM     128
#define NVEC    3

// Tiling
#define NTHREADS 768          // 24 waves (wave32): 8 M-tiles x 3 N-tiles
#define NCOLS    48           // columns per workgroup GEMM panel
#define BPAD     132          // padded B row stride (floats): conflict-free b64 LDS loads
#define OPITCH   388          // padded per-batch stride in output staging (384+4)
#define T_PER_WG 16           // atoms per workgroup in the output kernel

typedef __attribute__((ext_vector_type(2))) float v2f;
typedef __attribute__((ext_vector_type(8))) float v8f;

// ---------------------------------------------------------------------------
// CDNA5 async global->LDS copy (16B per lane), tracked by ASYNCcnt.
// LDS destination address taken from the vdst VGPR; per-lane global address in a VGPR pair.
// ---------------------------------------------------------------------------
__device__ __forceinline__ void async_copy_b128(uint32_t lds_off, const float* gptr) {
  asm volatile("global_load_async_to_lds_b128 %0, %1, off"
               :: "v"(lds_off), "v"((uint64_t)(uintptr_t)gptr)
               : "memory");
}
__device__ __forceinline__ void wait_asynccnt_le2() {
  asm volatile("s_wait_asynccnt 0x2" ::: "memory");
}

// ---------------------------------------------------------------------------
// Preload all 32 A-fragments (16x4 f32) for this wave's M-tile into regs.
// 32-bit A 16x4 layout: lane L<16 -> row M=L, k={0,1}; lane>=16 -> k={2,3}.
// ---------------------------------------------------------------------------
__device__ __forceinline__ void load_a_frags(const float* __restrict__ A,
                                             int mt, int lane, v2f* af) {
  const int m   = mt * 16 + (lane & 15);
  const int klo = (lane < 16) ? 0 : 2;
  const float* row = A + m * DIM + klo;
  #pragma unroll
  for (int s = 0; s < 32; ++s)
    af[s] = *reinterpret_cast<const v2f*>(row + 4 * s);
}

// Plain K=128 accumulation (B panel in LDS, BPAD pitch).
__device__ __forceinline__ v8f wmma_acc(const v2f* af, const float* Blds,
                                        int nt, int lane) {
  const int col = nt * 16 + (lane & 15);
  const int klo = (lane < 16) ? 0 : 2;
  const float* bp = Blds + col * BPAD + klo;
  v8f c = {};
  #pragma unroll
  for (int s = 0; s < 32; ++s) {
    v2f b = *reinterpret_cast<const v2f*>(bp + 4 * s);
    c = __builtin_amdgcn_wmma_f32_16x16x4_f32(false, af[s], false, b,
                                              (short)0, c, false, false);
  }
  return c;
}

// K-diagonal-scaled accumulation: C = (A * diag(m)) * B.  Lets the B panel be
// a raw async copy of H; mod[t] is a 128-float broadcast vector in LDS.
__device__ __forceinline__ v8f wmma_acc_scaled(const v2f* af, const float* Blds,
                                               const float* Mlds, int nt, int lane) {
  const int col = nt * 16 + (lane & 15);
  const int klo = (lane < 16) ? 0 : 2;
  const float* bp = Blds + col * BPAD + klo;
  const float* mp = Mlds + klo;
  v8f c = {};
  #pragma unroll
  for (int s = 0; s < 32; ++s) {
    v2f b = *reinterpret_cast<const v2f*>(bp + 4 * s);
    v2f m = *reinterpret_cast<const v2f*>(mp + 4 * s);
    v2f a = af[s] * m;
    c = __builtin_amdgcn_wmma_f32_16x16x4_f32(false, a, false, b,
                                              (short)0, c, false, false);
  }
  return c;
}

// ---------------------------------------------------------------------------
// Kernel 1: mod[t,f] = sum_e emb[t,e] * W_emb[f,e] + b_emb[f]
// ---------------------------------------------------------------------------
__global__ __launch_bounds__(NTHREADS) void mod_kernel(
    const float* __restrict__ emb, const float* __restrict__ W_emb,
    const float* __restrict__ b_emb, float* __restrict__ mod) {
  __shared__ __align__(16) float Blds[NCOLS * BPAD];
  const int tid = threadIdx.x, lane = tid & 31, wave = tid >> 5;
  const int mt = wave / 3, nt = wave % 3;
  v2f af[32];
  load_a_frags(W_emb, mt, lane, af);

  const int t0 = blockIdx.x * NCOLS;
  {
    const int col = tid >> 4, e0 = (tid & 15) * 8;
    const int t = min(t0 + col, ATOMS - 1);
    const float4 h0 = *reinterpret_cast<const float4*>(emb + (size_t)t * DIM + e0);
    const float4 h1 = *reinterpret_cast<const float4*>(emb + (size_t)t * DIM + e0 + 4);
    *reinterpret_cast<float4*>(&Blds[col * BPAD + e0])     = h0;
    *reinterpret_cast<float4*>(&Blds[col * BPAD + e0 + 4]) = h1;
  }
  __syncthreads();
  v8f c = wmma_acc(af, Blds, nt, lane);

  const int frow = mt * 16 + ((lane < 16) ? 0 : 8);
  const int t = t0 + nt * 16 + (lane & 15);
  if (t < ATOMS) {
    const float4 bb0 = *reinterpret_cast<const float4*>(b_emb + frow);
    const float4 bb1 = *reinterpret_cast<const float4*>(b_emb + frow + 4);
    float4 o0 = make_float4(c[0] + bb0.x, c[1] + bb0.y, c[2] + bb0.z, c[3] + bb0.w);
    float4 o1 = make_float4(c[4] + bb1.x, c[5] + bb1.y, c[6] + bb1.z, c[7] + bb1.w);
    *reinterpret_cast<float4*>(mod + (size_t)t * DIM + frow)     = o0;
    *reinterpret_cast<float4*>(mod + (size_t)t * DIM + frow + 4) = o1;
  }
}

// ---------------------------------------------------------------------------
// Kernel 2: H[b][a][v][e] = sum_c x[b][a][c][v] * W_in[e][c]
// Stored e-contiguous so the output kernel's gather is a contiguous copy.
// ---------------------------------------------------------------------------
__global__ __launch_bounds__(NTHREADS) void hproj_kernel(
    const float* __restrict__ x, const float* __restrict__ W_in,
    float* __restrict__ H) {
  __shared__ __align__(16) float Blds[NCOLS * BPAD];
  const int tid = threadIdx.x, lane = tid & 31, wave = tid >> 5;
  const int mt = wave / 3, nt = wave % 3;
  v2f af[32];
  load_a_frags(W_in, mt, lane, af);

  const int b  = blockIdx.x >> 6;
  const int a0 = (blockIdx.x & 63) * 16;
  const float* xblk = x + (size_t)(b * AMINOS + a0) * (DIM * NVEC);
  #pragma unroll
  for (int i = 0; i < 8; ++i) {
    const int f   = tid + i * NTHREADS;   // coalesced read, transposed LDS write
    const float val = xblk[f];
    const int a_l = f / (DIM * NVEC);
    const int r   = f - a_l * (DIM * NVEC);
    const int cch = r / NVEC;
    const int v   = r - cch * NVEC;
    Blds[(a_l * NVEC + v) * BPAD + cch] = val;
  }
  __syncthreads();
  v8f c = wmma_acc(af, Blds, nt, lane);

  const int col  = nt * 16 + (lane & 15);
  const int a_l  = col / NVEC, v = col - a_l * NVEC;
  const int erow = mt * 16 + ((lane < 16) ? 0 : 8);
  float* dst = H + (size_t)((b * AMINOS + a0 + a_l) * NVEC + v) * DIM + erow;
  *reinterpret_cast<float4*>(dst)     = make_float4(c[0], c[1], c[2], c[3]);
  *reinterpret_cast<float4*>(dst + 4) = make_float4(c[4], c[5], c[6], c[7]);
}

// ---------------------------------------------------------------------------
// Kernel 3 (dominant): out[b][t][d][v] =
//   sum_e W_out[d][e] * mod[t][e] * H[b][amino_idx[t]][v][e]
// Double-buffered async global->LDS gather of H (ASYNCcnt), mod folded into
// the register-resident A fragments, C staged through padded LDS for
// perfectly coalesced float4 stores of the 201 MB output.
// ---------------------------------------------------------------------------
__global__ __launch_bounds__(NTHREADS) void out_kernel(
    const float* __restrict__ H, const float* __restrict__ mod,
    const int* __restrict__ amino_idx, const float* __restrict__ W_out,
    float* __restrict__ out) {
  __shared__ __align__(16) float Hbuf[2][NCOLS * BPAD];
  __shared__ __align__(16) float Mlds[2][DIM];
  __shared__ __align__(16) float Olds[BATCH * OPITCH];
  const int tid = threadIdx.x, lane = tid & 31, wave = tid >> 5;
  const int mt = wave / 3, nt = wave % 3;
  v2f af[32];
  load_a_frags(W_out, mt, lane, af);

  // this thread's 32B slot in the gather panel: 48 columns = (b, v), 8 e's
  const int fcol = tid >> 4, e0 = (tid & 15) * 8;
  const int fb = fcol / NVEC, fv = fcol - fb * NVEC;

  uint32_t ldsHoff[2];  // wave-relative LDS byte offsets (flat addr low 32b)
  ldsHoff[0] = (uint32_t)(uintptr_t)&Hbuf[0][fcol * BPAD + e0];
  ldsHoff[1] = (uint32_t)(uintptr_t)&Hbuf[1][fcol * BPAD + e0];

  const int tbeg = blockIdx.x * T_PER_WG;
  const int tend = tbeg + T_PER_WG;

  auto issue_fill = [&](int t, int pp) {
    const int aidx = amino_idx[t];
    const float* hp = H + (size_t)((fb * AMINOS + aidx) * NVEC + fv) * DIM + e0;
    async_copy_b128(ldsHoff[pp], hp);
    async_copy_b128(ldsHoff[pp] + 16, hp + 4);
    if (tid < 32) {  // stage mod[t] broadcast vector
      *reinterpret_cast<float4*>(&Mlds[pp][tid * 4]) =
          *reinterpret_cast<const float4*>(mod + (size_t)t * DIM + tid * 4);
    }
  };

  issue_fill(tbeg, 0);
  int p = 0;
  for (int t = tbeg; t < tend; ++t) {
    const int tn = (t + 1 < tend) ? t + 1 : t;
    issue_fill(tn, p ^ 1);          // prefetch next atom's panel (async)
    wait_asynccnt_le2();            // current panel's copies complete
    __syncthreads();                // visible to all waves; Olds reads done

    v8f c = wmma_acc_scaled(af, &Hbuf[p][0], &Mlds[p][0], nt, lane);

    {  // stage C tile: Olds[b][d][v], padded pitch
      const int col  = nt * 16 + (lane & 15);
      const int cb   = col / NVEC, cv = col - cb * NVEC;
      const int drow = mt * 16 + ((lane < 16) ? 0 : 8);
      float* op = &Olds[cb * OPITCH + drow * NVEC + cv];
      #pragma unroll
      for (int j = 0; j < 8; ++j) op[j * NVEC] = c[j];
    }
    __syncthreads();

    // coalesced float4 output stores: out[((b*ATOMS+t)*DIM+d)*3+v]
    #pragma unroll
    for (int i = 0; i < 2; ++i) {
      const int q  = tid + i * NTHREADS;          // float4 index 0..1535
      const int ob = q / 96;                      // 96 float4 per batch row
      const int r4 = q - ob * 96;
      const float4 val = *reinterpret_cast<const float4*>(&Olds[ob * OPITCH + r4 * 4]);
      *reinterpret_cast<float4*>(out + (size_t)(ob * ATOMS + t) * (DIM * NVEC) + r4 * 4) = val;
    }
    p ^= 1;
  }
}

// ---------------------------------------------------------------------------
extern "C" void kernel_launch(void* const* d_in, const int* in_sizes, int n_in,
                              void* d_out, int out_size, void* d_ws, size_t ws_size,
                              hipStream_t stream) {
  const float* x     = (const float*)d_in[0];
  const float* emb   = (const float*)d_in[1];
  const int*   aidx  = (const int*)d_in[2];
  const float* W_emb = (const float*)d_in[3];
  const float* b_emb = (const float*)d_in[4];
  const float* W_in  = (const float*)d_in[5];
  const float* W_out = (const float*)d_in[6];
  float* out = (float*)d_out;

  // workspace: H (25.2 MB) then mod (4.2 MB)
  float* H   = (float*)d_ws;
  float* mod = (float*)((char*)d_ws + (size_t)BATCH * AMINOS * NVEC * DIM * sizeof(float));

  mod_kernel  <<<(ATOMS + NCOLS - 1) / NCOLS, NTHREADS, 0, stream>>>(emb, W_emb, b_emb, mod);
  hproj_kernel<<<BATCH * (AMINOS / 16),       NTHREADS, 0, stream>>>(x, W_in, H);
  out_kernel  <<<ATOMS / T_PER_WG,            NTHREADS, 0, stream>>>(H, mod, aidx, W_out, out);
}